// DinoV3Attention_41300405518720
// MI455X (gfx1250) — compile-verified
//
#include <hip/hip_runtime.h>

// ---------------------------------------------------------------------------
// DinoV3 attention for MI455X (gfx1250, wave32, WMMA + async-to-LDS).
// fp32 end-to-end with V_WMMA_F32_16X16X4_F32 (exact fp32 matrix math).
// ~138 GFLOP vs ~200MB HBM traffic -> memory-bound even on the fp32 matrix
// pipe, so exact fp32 WMMA is the right precision choice.
//
// All tiles are staged with GLOBAL_LOAD_ASYNC_TO_LDS_B128 (ASYNCcnt) into
// double-buffered LDS; the contraction dimension is made contiguous in
// global memory beforehand (weights pre-transposed once, V stored [B,H,d,s])
// so every WMMA fragment is a single aligned 8-byte LDS load.
//
// Workspace (floats): [Q | K | Vt | attn_out | WtQ WtK WtV WtO] ~= 152 MB.
// ---------------------------------------------------------------------------

typedef __attribute__((ext_vector_type(2))) float v2f;
typedef __attribute__((ext_vector_type(8))) float v8f;

#define B_   8
#define S_   1029
#define SP_  1032               // padded V^T row stride (16B-aligned tiles)
#define D_   1024
#define H_   16
#define HD_  64
#define NPRE 5
#define M_   (B_ * S_)          // 8232
#define QSCALE 0.125f           // 64^-0.5

// D = A(16x4) * B(4x16) + C   (fp32 WMMA, K-step = 4)
// A-frag: lane l holds M=l%16, K = 2*(l/16)+{0,1}
// B-frag: lane l holds N=l%16, K = 2*(l/16)+{0,1}
// C/D   : lane l holds N=l%16, M = r + 8*(l/16) in VGPR r
#define WMMA4(a, b, c) \
  __builtin_amdgcn_wmma_f32_16x16x4_f32(false, (a), false, (b), (short)0, (c), false, false)

// Async copy: 16 bytes global -> LDS per lane, tracked by ASYNCcnt.
__device__ __forceinline__ void async_b128(void* lds_ptr, const void* gptr) {
  unsigned int       loff = (unsigned int)(size_t)lds_ptr;   // low32 = LDS offset
  unsigned long long ga   = (unsigned long long)(size_t)gptr;
  asm volatile("global_load_async_to_lds_b128 %0, %1, off"
               :: "v"(loff), "v"(ga) : "memory");
}
__device__ __forceinline__ void wait_async0() {
  asm volatile("s_wait_asynccnt 0x0" ::: "memory");
}

// ---------------------------------------------------------------------------
// 1024x1024 transpose (for weights): out[n][k] = in[k][n]
// ---------------------------------------------------------------------------
__global__ __launch_bounds__(256) void transpose1024_kernel(
    const float* __restrict__ in, float* __restrict__ out)
{
  __shared__ float t[64][65];
  const int bx = blockIdx.x * 64;   // col tile
  const int by = blockIdx.y * 64;   // row tile
  #pragma unroll
  for (int i = 0; i < 4; ++i) {
    int id = threadIdx.x + 256 * i;          // 0..1023
    int r  = id >> 4;
    int c4 = (id & 15) * 4;
    float4 v = *reinterpret_cast<const float4*>(&in[(size_t)(by + r) * 1024 + bx + c4]);
    t[r][c4 + 0] = v.x; t[r][c4 + 1] = v.y; t[r][c4 + 2] = v.z; t[r][c4 + 3] = v.w;
  }
  __syncthreads();
  #pragma unroll
  for (int i = 0; i < 4; ++i) {
    int id = threadIdx.x + 256 * i;
    int r  = id >> 4;
    int c4 = (id & 15) * 4;
    float4 v;
    v.x = t[c4 + 0][r]; v.y = t[c4 + 1][r]; v.z = t[c4 + 2][r]; v.w = t[c4 + 3][r];
    *reinterpret_cast<float4*>(&out[(size_t)(bx + r) * 1024 + by + c4]) = v;
  }
}

// Zero the 3 pad columns of every V^T row (so masked cols are finite zeros).
__global__ void vpad_kernel(float* __restrict__ vt) {
  int idx = blockIdx.x * blockDim.x + threadIdx.x;   // B*H*64*3
  if (idx >= B_ * H_ * HD_ * 3) return;
  int p = idx % 3;
  int row = idx / 3;
  vt[(size_t)row * SP_ + S_ + p] = 0.f;
}

// ---------------------------------------------------------------------------
// GEMM: C[M x 1024] = A[M x 1024] @ W[1024 x 1024] (+ bias), W given TRANSPOSED.
// 256 thr (8 waves), block tile 128(M) x 64(N), K chunks of 32, async double
// buffering. mode: 0 = flat [M,1024]; 1 = head scatter [B,H,S,64];
//                  2 = transposed head scatter [B,H,64,SP_] (for V).
// ---------------------------------------------------------------------------
__global__ __launch_bounds__(256) void gemm1024_kernel(
    const float* __restrict__ A, const float* __restrict__ Wt,
    const float* __restrict__ bias, float* __restrict__ Cout,
    int M, int mode)
{
  __shared__ float Ab[2][128][36];   // [m][k]; 144B rows (16B aligned)
  __shared__ float Wb[2][64][36];    // [n][k]; 144B rows (16B aligned)

  const int tid  = threadIdx.x;
  const int wave = tid >> 5;
  const int lane = tid & 31;
  const int l16  = lane & 15;
  const int hi   = lane >> 4;
  const int m0   = blockIdx.y * 128;
  const int n0   = blockIdx.x * 64;

  auto issue = [&](int kc, int bufi) {
    const int k0 = kc * 32;
    #pragma unroll
    for (int i = 0; i < 4; ++i) {                  // A tile 128x32
      int id  = tid + 256 * i;
      int row = id >> 3;
      int c4  = (id & 7) * 4;
      int gm  = m0 + row;
      if (gm < M)                                  // OOB rows: store-guarded
        async_b128(&Ab[bufi][row][c4], &A[(size_t)gm * 1024 + k0 + c4]);
    }
    #pragma unroll
    for (int i = 0; i < 2; ++i) {                  // W^T tile 64(n) x 32(k)
      int id  = tid + 256 * i;
      int row = id >> 3;
      int c4  = (id & 7) * 4;
      async_b128(&Wb[bufi][row][c4], &Wt[(size_t)(n0 + row) * 1024 + k0 + c4]);
    }
  };

  v8f acc[4] = {};
  issue(0, 0);

  for (int kc = 0; kc < 32; ++kc) {
    const int cur = kc & 1;
    wait_async0();
    __syncthreads();                               // tile kc ready; prev reads done
    if (kc + 1 < 32) issue(kc + 1, 1 - cur);       // prefetch overlaps WMMA

    const int arow = wave * 16 + l16;
    #pragma unroll
    for (int t = 0; t < 8; ++t) {
      v2f af = *(const v2f*)(&Ab[cur][arow][4 * t + 2 * hi]);
      #pragma unroll
      for (int ns = 0; ns < 4; ++ns) {
        v2f bf = *(const v2f*)(&Wb[cur][ns * 16 + l16][4 * t + 2 * hi]);
        acc[ns] = WMMA4(af, bf, acc[ns]);
      }
    }
  }

  #pragma unroll
  for (int ns = 0; ns < 4; ++ns) {
    int n = n0 + ns * 16 + l16;
    float bv = bias ? bias[n] : 0.f;
    #pragma unroll
    for (int r = 0; r < 8; ++r) {
      int m = m0 + wave * 16 + r + 8 * hi;
      if (m < M) {
        float val = acc[ns][r] + bv;
        int b = m / S_; int s = m - b * S_;
        int h = n >> 6; int d = n & 63;
        if (mode == 1)
          Cout[(((size_t)(b * H_ + h)) * S_ + s) * HD_ + d] = val;
        else if (mode == 2)
          Cout[(((size_t)(b * H_ + h)) * HD_ + d) * SP_ + s] = val;
        else
          Cout[(size_t)m * 1024 + n] = val;
      }
    }
  }
}

// ---------------------------------------------------------------------------
// RoPE (+ Q scale) in-place on Q,K in [B,H,S,64]; thread owns a (d,d+32) pair.
// ---------------------------------------------------------------------------
__global__ __launch_bounds__(256) void rope_kernel(
    float* __restrict__ q, float* __restrict__ k,
    const float* __restrict__ cosp, const float* __restrict__ sinp)
{
  int idx = blockIdx.x * blockDim.x + threadIdx.x;   // B*H*S*32
  if (idx >= B_ * H_ * S_ * 32) return;
  int j  = idx & 31;
  int s  = (idx >> 5) % S_;
  int bh = idx / (S_ * 32);
  float* qp = q + ((size_t)bh * S_ + s) * HD_;
  float* kp = k + ((size_t)bh * S_ + s) * HD_;
  if (s >= NPRE) {
    int p = s - NPRE;
    float c1 = cosp[p * HD_ + j],      s1 = sinp[p * HD_ + j];
    float c2 = cosp[p * HD_ + j + 32], s2 = sinp[p * HD_ + j + 32];
    float q1 = qp[j], q2 = qp[j + 32];
    qp[j]      = (q1 * c1 - q2 * s1) * QSCALE;
    qp[j + 32] = (q2 * c2 + q1 * s2) * QSCALE;
    float k1 = kp[j], k2 = kp[j + 32];
    kp[j]      = k1 * c1 - k2 * s1;
    kp[j + 32] = k2 * c2 + k1 * s2;
  } else {
    qp[j]      *= QSCALE;
    qp[j + 32] *= QSCALE;
  }
}

// ---------------------------------------------------------------------------
// Flash attention per (b,h): block = 128 q rows (8 waves x 16 rows).
// 32-col K/V tiles, async double buffering, online softmax per wave,
// P through wave-private LDS. V comes pre-transposed ([B,H,d,s]).
// ---------------------------------------------------------------------------
__global__ __launch_bounds__(256) void attn_kernel(
    const float* __restrict__ Q, const float* __restrict__ K,
    const float* __restrict__ Vt, float* __restrict__ Out)
{
  __shared__ float Kb[2][32][68];      // [k][d], 272B rows (16B aligned)
  __shared__ float Vb[2][64][36];      // [d][k], 144B rows (16B aligned)
  __shared__ float Plds[8][16][34];    // per-wave P [qrow][kcol]

  const int tid  = threadIdx.x;
  const int wave = tid >> 5;
  const int lane = tid & 31;
  const int l16  = lane & 15;
  const int hi   = lane >> 4;

  const int qt = blockIdx.x;
  const int h  = blockIdx.y;
  const int b  = blockIdx.z;
  const float* Qh  = Q  + ((size_t)(b * H_ + h)) * S_ * HD_;
  const float* Kh  = K  + ((size_t)(b * H_ + h)) * S_ * HD_;
  const float* Vth = Vt + ((size_t)(b * H_ + h)) * HD_ * SP_;

  auto issue = [&](int kt, int bufi) {
    const int kbase = kt * 32;
    #pragma unroll
    for (int i = 0; i < 2; ++i) {                  // K tile 32(k) x 64(d)
      int id  = tid + 256 * i;
      int row = id >> 4;
      int c4  = (id & 15) * 4;
      int gk  = kbase + row;
      if (gk < S_)                                 // stale rows masked to -inf
        async_b128(&Kb[bufi][row][c4], Kh + (size_t)gk * HD_ + c4);
    }
    #pragma unroll
    for (int i = 0; i < 2; ++i) {                  // V^T tile 64(d) x 32(s)
      int id  = tid + 256 * i;
      int row = id >> 3;
      int c4  = (id & 7) * 4;
      if (kbase + c4 < S_)   // chunk may touch zeroed pads; never past row end
        async_b128(&Vb[bufi][row][c4], Vth + (size_t)row * SP_ + kbase + c4);
    }
  };

  // Q A-fragments in registers for the whole kv loop
  const int qrow = qt * 128 + wave * 16 + l16;
  const int qrc  = qrow < S_ ? qrow : (S_ - 1);
  v2f qf[16];
  #pragma unroll
  for (int t = 0; t < 16; ++t)
    qf[t] = *(const v2f*)(Qh + (size_t)qrc * HD_ + 4 * t + 2 * hi);

  float mrun[8], lrun[8];
  v8f o[4] = {};
  #pragma unroll
  for (int r = 0; r < 8; ++r) { mrun[r] = -__builtin_inff(); lrun[r] = 0.f; }

  const int NKT = (S_ + 31) / 32;        // 33
  issue(0, 0);

  for (int kt = 0; kt < NKT; ++kt) {
    const int cur   = kt & 1;
    const int kbase = kt * 32;
    wait_async0();
    __syncthreads();                     // tile kt visible; prev reads done
    if (kt + 1 < NKT) issue(kt + 1, 1 - cur);

    // ---- scores: 16 q-rows x 32 k-cols per wave ----
    v8f sc[2] = {};
    #pragma unroll
    for (int t = 0; t < 16; ++t) {
      #pragma unroll
      for (int ns = 0; ns < 2; ++ns) {
        v2f bf = *(const v2f*)(&Kb[cur][ns * 16 + l16][4 * t + 2 * hi]);
        sc[ns] = WMMA4(qf[t], bf, sc[ns]);
      }
    }
    #pragma unroll
    for (int ns = 0; ns < 2; ++ns) {
      int col = kbase + ns * 16 + l16;
      if (col >= S_) {
        #pragma unroll
        for (int r = 0; r < 8; ++r) sc[ns][r] = -__builtin_inff();
      }
    }

    // ---- online softmax (row = r + 8*hi; cols across half-wave) ----
    float alpha[8];
    #pragma unroll
    for (int r = 0; r < 8; ++r) {
      float v = fmaxf(sc[0][r], sc[1][r]);
      #pragma unroll
      for (int off = 8; off >= 1; off >>= 1)
        v = fmaxf(v, __shfl_xor(v, off, 32));
      float mnew = fmaxf(mrun[r], v);
      alpha[r] = __expf(mrun[r] - mnew);
      mrun[r] = mnew;
      float srow = 0.f;
      #pragma unroll
      for (int ns = 0; ns < 2; ++ns) {
        float p = __expf(sc[ns][r] - mnew);
        Plds[wave][r + 8 * hi][ns * 16 + l16] = p;   // wave-private, in-order
        srow += p;
      }
      #pragma unroll
      for (int off = 8; off >= 1; off >>= 1)
        srow += __shfl_xor(srow, off, 32);
      lrun[r] = lrun[r] * alpha[r] + srow;
    }
    #pragma unroll
    for (int ns = 0; ns < 4; ++ns)
      #pragma unroll
      for (int r = 0; r < 8; ++r) o[ns][r] *= alpha[r];

    // ---- O += P(16x32) @ V(32x64), V^T tile has contiguous k ----
    #pragma unroll
    for (int t = 0; t < 8; ++t) {
      v2f pf = *(const v2f*)(&Plds[wave][l16][4 * t + 2 * hi]);
      #pragma unroll
      for (int ns = 0; ns < 4; ++ns) {
        v2f vf = *(const v2f*)(&Vb[cur][ns * 16 + l16][4 * t + 2 * hi]);
        o[ns] = WMMA4(pf, vf, o[ns]);
      }
    }
  }

  // ---- normalize + store [B,S,D] ----
  #pragma unroll
  for (int ns = 0; ns < 4; ++ns) {
    int d = ns * 16 + l16;
    #pragma unroll
    for (int r = 0; r < 8; ++r) {
      int m = qt * 128 + wave * 16 + r + 8 * hi;
      if (m < S_)
        Out[((size_t)(b * S_ + m)) * D_ + h * HD_ + d] = o[ns][r] / lrun[r];
    }
  }
}

// ---------------------------------------------------------------------------
extern "C" void kernel_launch(void* const* d_in, const int* in_sizes, int n_in,
                              void* d_out, int out_size, void* d_ws, size_t ws_size,
                              hipStream_t stream) {
  const float* x        = (const float*)d_in[0];
  const float* rope_cos = (const float*)d_in[1];
  const float* rope_sin = (const float*)d_in[2];
  const float* Wq = (const float*)d_in[3];
  const float* bq = (const float*)d_in[4];
  const float* Wk = (const float*)d_in[5];
  const float* Wv = (const float*)d_in[6];
  const float* bv = (const float*)d_in[7];
  const float* Wo = (const float*)d_in[8];
  const float* bo = (const float*)d_in[9];
  float* out = (float*)d_out;

  float* ws = (float*)d_ws;
  const size_t perQK = (size_t)B_ * H_ * S_ * HD_;   // 8,429,568
  const size_t perVT = (size_t)B_ * H_ * HD_ * SP_;  // 8,454,144
  float* qws  = ws;
  float* kws  = qws + perQK;
  float* vtws = kws + perQK;
  float* aout = vtws + perVT;
  float* wtq  = aout + perQK;
  float* wtk  = wtq + (size_t)1024 * 1024;
  float* wtv  = wtk + (size_t)1024 * 1024;
  float* wto  = wtv + (size_t)1024 * 1024;

  dim3 blk(256);

  // one-time weight transposes (amortized across 65 M-blocks per GEMM)
  dim3 tgrid(16, 16);
  transpose1024_kernel<<<tgrid, blk, 0, stream>>>(Wq, wtq);
  transpose1024_kernel<<<tgrid, blk, 0, stream>>>(Wk, wtk);
  transpose1024_kernel<<<tgrid, blk, 0, stream>>>(Wv, wtv);
  transpose1024_kernel<<<tgrid, blk, 0, stream>>>(Wo, wto);
  vpad_kernel<<<(B_ * H_ * HD_ * 3 + 255) / 256, blk, 0, stream>>>(vtws);

  dim3 ggrid(1024 / 64, (M_ + 127) / 128);           // (16, 65)
  gemm1024_kernel<<<ggrid, blk, 0, stream>>>(x, wtq, bq,      qws,  M_, 1);
  gemm1024_kernel<<<ggrid, blk, 0, stream>>>(x, wtk, nullptr, kws,  M_, 1);
  gemm1024_kernel<<<ggrid, blk, 0, stream>>>(x, wtv, bv,      vtws, M_, 2);

  int nrope = B_ * H_ * S_ * 32;
  rope_kernel<<<(nrope + 255) / 256, blk, 0, stream>>>(qws, kws, rope_cos, rope_sin);

  dim3 agrid((S_ + 127) / 128, H_, B_);              // (9, 16, 8)
  attn_kernel<<<agrid, blk, 0, stream>>>(qws, kws, vtws, aout);

  gemm1024_kernel<<<ggrid, blk, 0, stream>>>(aout, wto, bo, out, M_, 0);
}